// Pooling_fps_69604239999565
// MI455X (gfx1250) — compile-verified
//
#include <hip/hip_runtime.h>
#include <cstdint>

// Problem constants (match reference)
#define BB 8
#define NN 4096
#define CC 128
#define KK 16
#define NPOINT 1024

// ---------------------------------------------------------------------------
// CDNA5 async global->LDS helpers (ASYNCcnt-tracked data mover).
// Builtin prototypes (probed via compiler diagnostics):
//   __builtin_amdgcn_global_load_async_to_lds_b32 (global int*,  lds int*,  imm, imm)
//   __builtin_amdgcn_global_load_async_to_lds_b128(global int4*, lds int4*, imm, imm)
// ---------------------------------------------------------------------------
#if defined(__HIP_DEVICE_COMPILE__) &&                                        \
    __has_builtin(__builtin_amdgcn_global_load_async_to_lds_b128) &&          \
    __has_builtin(__builtin_amdgcn_global_load_async_to_lds_b32)
#define USE_ASYNC_LDS 1
#else
#define USE_ASYNC_LDS 0
#endif

#if USE_ASYNC_LDS
typedef int v4i_t __attribute__((ext_vector_type(4)));
typedef __attribute__((address_space(1))) v4i_t* g_v4i_ptr;
typedef __attribute__((address_space(3))) v4i_t* l_v4i_ptr;
typedef __attribute__((address_space(1))) int*   g_i32_ptr;
typedef __attribute__((address_space(3))) int*   l_i32_ptr;
#endif

__device__ __forceinline__ void async_g2l_b128(void* lds_dst, const void* g_src) {
#if USE_ASYNC_LDS
  // as(1) ptr = full 64-bit flat addr; as(3) ptr = low 32 bits (LDS offset).
  __builtin_amdgcn_global_load_async_to_lds_b128(
      (g_v4i_ptr)(unsigned long long)(uintptr_t)g_src,
      (l_v4i_ptr)(unsigned)(uintptr_t)lds_dst,
      0, 0);
#else
  *(float4*)lds_dst = *(const float4*)g_src;
#endif
}

__device__ __forceinline__ void async_g2l_b32(void* lds_dst, const void* g_src) {
#if USE_ASYNC_LDS
  __builtin_amdgcn_global_load_async_to_lds_b32(
      (g_i32_ptr)(unsigned long long)(uintptr_t)g_src,
      (l_i32_ptr)(unsigned)(uintptr_t)lds_dst,
      0, 0);
#else
  *(float*)lds_dst = *(const float*)g_src;
#endif
}

__device__ __forceinline__ void async_wait_all() {
#if USE_ASYNC_LDS
#if __has_builtin(__builtin_amdgcn_s_wait_asynccnt)
  __builtin_amdgcn_s_wait_asynccnt(0);
#else
  asm volatile("s_wait_asynccnt 0" ::: "memory");
#endif
#endif
}

// ---------------------------------------------------------------------------
// Kernel 1: farthest point sampling. One block (256 threads = 8 wave32) per
// batch. xyz staged to LDS via async data mover; per-thread distances live in
// VGPRs (16 points/thread, strided). Argmax: wave shfl_xor reduction + tiny
// LDS cross-wave step. Writes centroid indices (ws) and vertices_pool (d_out).
// Matches jax scan semantics: cents[0]=0; cents[i]=argmax after i updates;
// ties -> lowest index (jnp.argmax first occurrence).
// ---------------------------------------------------------------------------
__global__ __launch_bounds__(256) void fps_kernel(
    const float* __restrict__ vertices,  // (B,3,N)
    float* __restrict__ out_v,           // (B,3,NPOINT)
    int* __restrict__ cents)             // (B,NPOINT)
{
  __shared__ alignas(16) float xs[NN];
  __shared__ alignas(16) float ys[NN];
  __shared__ alignas(16) float zs[NN];
  __shared__ float redV[8];
  __shared__ int   redI[8];
  __shared__ float wcx, wcy, wcz;
  __shared__ int   widx;

  const int b   = blockIdx.x;
  const int tid = threadIdx.x;
  const int lane = tid & 31;
  const int wv   = tid >> 5;

  // Stage xyz (3 rows x 16KB) into LDS with the async data mover.
  const float4* gx4 = (const float4*)(vertices + (size_t)b * 3 * NN);
  const float4* gy4 = gx4 + NN / 4;
  const float4* gz4 = gy4 + NN / 4;
  float4* xs4 = (float4*)xs;
  float4* ys4 = (float4*)ys;
  float4* zs4 = (float4*)zs;
#pragma unroll
  for (int r = 0; r < 4; ++r) {
    int i = tid + r * 256;  // float4 index, 1024 per row
    async_g2l_b128(&xs4[i], &gx4[i]);
    async_g2l_b128(&ys4[i], &gy4[i]);
    async_g2l_b128(&zs4[i], &gz4[i]);
  }
  async_wait_all();
  __syncthreads();

  // Cache this thread's 16 points (strided) + distances in registers.
  float lx[16], ly[16], lz[16], dd[16];
#pragma unroll
  for (int j = 0; j < 16; ++j) {
    int p = tid + (j << 8);
    lx[j] = xs[p];
    ly[j] = ys[p];
    lz[j] = zs[p];
    dd[j] = 1e10f;
  }

  int   cur = 0;
  float ccx = xs[0], ccy = ys[0], ccz = zs[0];

  for (int it = 0; it < NPOINT; ++it) {
    if (tid == 0) {
      cents[b * NPOINT + it] = cur;
      out_v[(size_t)b * 3 * NPOINT + it]              = ccx;
      out_v[(size_t)b * 3 * NPOINT + NPOINT + it]     = ccy;
      out_v[(size_t)b * 3 * NPOINT + 2 * NPOINT + it] = ccz;
    }
    if (it == NPOINT - 1) break;

    float bestV = -1.0f;
    int   bestI = 0x7fffffff;
#pragma unroll
    for (int j = 0; j < 16; ++j) {
      float dx = lx[j] - ccx;
      float dy = ly[j] - ccy;
      float dz = lz[j] - ccz;
      float dist = dx * dx + dy * dy + dz * dz;
      float nd = fminf(dd[j], dist);
      dd[j] = nd;
      int p = tid + (j << 8);
      if (nd > bestV || (nd == bestV && p < bestI)) { bestV = nd; bestI = p; }
    }
    // wave32 argmax reduction
#pragma unroll
    for (int off = 16; off > 0; off >>= 1) {
      float ov = __shfl_xor(bestV, off, 32);
      int   oi = __shfl_xor(bestI, off, 32);
      if (ov > bestV || (ov == bestV && oi < bestI)) { bestV = ov; bestI = oi; }
    }
    if (lane == 0) { redV[wv] = bestV; redI[wv] = bestI; }
    __syncthreads();
    if (tid == 0) {
      float bvv = redV[0];
      int   bii = redI[0];
#pragma unroll
      for (int w = 1; w < 8; ++w) {
        if (redV[w] > bvv || (redV[w] == bvv && redI[w] < bii)) {
          bvv = redV[w];
          bii = redI[w];
        }
      }
      widx = bii;
      wcx = xs[bii]; wcy = ys[bii]; wcz = zs[bii];
    }
    __syncthreads();
    cur = widx;
    ccx = wcx; ccy = wcy; ccz = wcz;
  }
}

// ---------------------------------------------------------------------------
// Kernel 2: transpose feature_map (B,C,N) -> flatT (B*N, C) row-major so that
// each KNN neighbor row is one contiguous 512B read. 32x32 tiles via LDS,
// loads issued through the async data mover.
// ---------------------------------------------------------------------------
__global__ __launch_bounds__(256) void transpose_kernel(
    const float* __restrict__ fm,   // (B,C,N)
    float* __restrict__ flatT)      // (B*N, C)
{
  __shared__ float tile[32][33];
  const int n0 = blockIdx.x * 32;
  const int c0 = blockIdx.y * 32;
  const int b  = blockIdx.z;
  const int tx = threadIdx.x;  // 0..31
  const int ty = threadIdx.y;  // 0..7

  const float* src = fm + ((size_t)b * CC + c0) * NN + n0;
#pragma unroll
  for (int r = 0; r < 4; ++r) {
    int cl = ty + 8 * r;
    async_g2l_b32(&tile[cl][tx], &src[(size_t)cl * NN + tx]);
  }
  async_wait_all();
  __syncthreads();

  float* dst = flatT + ((size_t)b * NN + n0) * CC + c0;
#pragma unroll
  for (int r = 0; r < 4; ++r) {
    int nl = ty + 8 * r;
    dst[(size_t)nl * CC + tx] = tile[tx][nl];
  }
}

// ---------------------------------------------------------------------------
// Kernel 3: gather-max-pool, computed ONLY at the 1024 centroids per batch
// (output equivalence: pooled is gathered at centroids). One wave per output
// point; each lane owns 4 channels (float4). Results staged in LDS then
// written out transposed (B,C,npoint) with s-contiguous coalesced stores.
// ---------------------------------------------------------------------------
__global__ __launch_bounds__(256) void gather_pool_kernel(
    const float* __restrict__ flatT,      // (B*N, C)
    const long long* __restrict__ idx,    // (B*N*K) values in [0, B*N)
    const int* __restrict__ cents,        // (B,NPOINT)
    float* __restrict__ out_f)            // (B,C,NPOINT)
{
  __shared__ alignas(16) float pooled[64][132];  // stride 132 -> low bank conflicts
  const int b   = blockIdx.y;
  const int s0  = blockIdx.x * 64;
  const int tid = threadIdx.x;
  const int wv  = tid >> 5;
  const int lane = tid & 31;

  for (int pt = wv; pt < 64; pt += 8) {
    const int s = s0 + pt;
    const int cent = cents[b * NPOINT + s];
    const long long base = ((long long)(b * NN + cent)) * KK;
    float4 m = make_float4(-3.402823466e38f, -3.402823466e38f,
                           -3.402823466e38f, -3.402823466e38f);
#pragma unroll 4
    for (int k = 0; k < KK; ++k) {
      long long j = idx[base + k];             // wave-uniform
      const float4* row = (const float4*)(flatT + j * CC);
      float4 v = row[lane];                    // contiguous 512B per wave
      m.x = fmaxf(m.x, v.x);
      m.y = fmaxf(m.y, v.y);
      m.z = fmaxf(m.z, v.z);
      m.w = fmaxf(m.w, v.w);
    }
    *(float4*)&pooled[pt][lane * 4] = m;
  }
  __syncthreads();

  float* outb = out_f + (size_t)b * CC * NPOINT + s0;
  for (int e = tid; e < 64 * CC; e += 256) {
    int c  = e >> 6;
    int sl = e & 63;
    outb[(size_t)c * NPOINT + sl] = pooled[sl][c];
  }
}

// ---------------------------------------------------------------------------
// Launch
// ---------------------------------------------------------------------------
extern "C" void kernel_launch(void* const* d_in, const int* in_sizes, int n_in,
                              void* d_out, int out_size, void* d_ws, size_t ws_size,
                              hipStream_t stream) {
  const float* vertices    = (const float*)d_in[0];        // (B,3,N)
  const float* feature_map = (const float*)d_in[1];        // (B,C,N)
  const long long* idx     = (const long long*)d_in[2];    // (B*N*K) int64

  float* out_v = (float*)d_out;                    // (B,3,NPOINT) = 24576 floats
  float* out_f = out_v + (size_t)BB * 3 * NPOINT;  // (B,C,NPOINT)

  int*   cents = (int*)d_ws;                       // 32 KB
  float* flatT = (float*)((char*)d_ws + 65536);    // 16 MB (B*N, C)

  fps_kernel<<<dim3(BB), dim3(256), 0, stream>>>(vertices, out_v, cents);
  transpose_kernel<<<dim3(NN / 32, CC / 32, BB), dim3(32, 8), 0, stream>>>(
      feature_map, flatT);
  gather_pool_kernel<<<dim3(NPOINT / 64, BB), dim3(256), 0, stream>>>(
      flatT, idx, cents, out_f);
}